// DRMMLogCountHistogram_5222680232145
// MI455X (gfx1250) — compile-verified
//
#include <hip/hip_runtime.h>
#include <stdint.h>

// DRMM log-count histogram, MI455X (gfx1250).
// simmat [B=128, Q=32, D=8192] f32, dtoks [B,D] i32, qtoks [B,Q] i32
// out    [B, Q, 30] f32 = log(weighted_bincount + 1e-5)
//
// Memory-bound: 134 MB of simmat @ 23.3 TB/s ~ 6 us floor. Use the CDNA5
// async data mover (GLOBAL_LOAD_ASYNC_TO_LDS_B128, ASYNCcnt) to stage each
// row into LDS, wave-private LDS histograms (ds_add_u32) for accumulation.

#define BINS        30
#define Q_DIM       32
#define D_DIM       8192
#define THREADS     256
#define NWAVE       8                 // 256 / wave32
#define CHUNK_ELEMS 1024              // 256 lanes * 4 floats
#define NCHUNK      (D_DIM / CHUNK_ELEMS)  // 8

// Per-lane async copy: 16 bytes global -> LDS. VDST = LDS byte offset (u32),
// VADDR = 64-bit global address, SADDR = off (GV mode). Tracked by ASYNCcnt.
__device__ __forceinline__ void async_load_b128(uint32_t lds_off, const void* gaddr) {
    asm volatile("global_load_async_to_lds_b128 %0, %1, off"
                 :
                 : "v"(lds_off), "v"(gaddr)
                 : "memory");
}

// Wait until at most `c` async ops outstanding (immediates must be literal).
__device__ __forceinline__ void wait_async_leq(int c) {
    switch (c) {
        case 0: asm volatile("s_wait_asynccnt 0" ::: "memory"); break;
        case 1: asm volatile("s_wait_asynccnt 1" ::: "memory"); break;
        case 2: asm volatile("s_wait_asynccnt 2" ::: "memory"); break;
        case 3: asm volatile("s_wait_asynccnt 3" ::: "memory"); break;
        case 4: asm volatile("s_wait_asynccnt 4" ::: "memory"); break;
        case 5: asm volatile("s_wait_asynccnt 5" ::: "memory"); break;
        case 6: asm volatile("s_wait_asynccnt 6" ::: "memory"); break;
        default: asm volatile("s_wait_asynccnt 7" ::: "memory"); break;
    }
}

__device__ __forceinline__ void accum_elem(unsigned* wave_hist, bool qw,
                                           float sim, int dtok) {
    if (qw & (dtok != 0)) {
        // match reference rounding: (s + 1.000001) / 2 * 29, truncating cast
        float x = (sim + 1.000001f) * 0.5f * 29.0f;
        int bin = (int)x;
        bin = bin < 0 ? 0 : (bin > 31 ? 31 : bin);   // stay inside 32-slot region
        atomicAdd(&wave_hist[bin], 1u);              // ds_add_u32 (no rtn)
    }
}

__global__ __launch_bounds__(THREADS)
void drmm_hist_kernel(const float* __restrict__ simmat,
                      const int*   __restrict__ dtoks,
                      const int*   __restrict__ qtoks,
                      float*       __restrict__ out) {
    __shared__ float    stage[D_DIM];        // 32 KB: one full simmat row
    __shared__ unsigned hcnt[NWAVE * 32];    // per-wave privatized histograms

    const int row  = blockIdx.x;             // 0 .. B*Q-1
    const int b    = row >> 5;               // row / Q_DIM (Q_DIM == 32)
    const int t    = threadIdx.x;
    const int wave = t >> 5;

    // zero histograms: each wave zeroes its own 32-slot region (lane j -> slot j),
    // and a wave's LDS ops are in-order, so no barrier needed before atomics.
    hcnt[t] = 0u;

    // Issue async staging of the whole row (8 x b128 per lane = 32 KB total).
    // Each lane later consumes exactly the bytes it copied -> per-wave
    // s_wait_asynccnt suffices, no workgroup barrier.
    const float*   grow = simmat + (size_t)row * D_DIM + (size_t)t * 4;
    const uint32_t lds0 = (uint32_t)(uintptr_t)(&stage[0]) + (uint32_t)t * 16u;
#pragma unroll
    for (int c = 0; c < NCHUNK; ++c) {
        async_load_b128(lds0 + (uint32_t)c * (CHUNK_ELEMS * 4u),
                        grow + (size_t)c * CHUNK_ELEMS);
    }

    const bool qw = (qtoks[row] != 0);       // qtoks[b*Q + q] == qtoks[row]
    const int* drow = dtoks + (size_t)b * D_DIM + (size_t)t * 4;
    unsigned*  wh   = &hcnt[wave * 32];

#pragma unroll
    for (int c = 0; c < NCHUNK; ++c) {
        // async loads complete in order: count <= NCHUNK-1-c  =>  chunk c landed
        wait_async_leq(NCHUNK - 1 - c);
        const float4 s  = *(const float4*)&stage[c * CHUNK_ELEMS + t * 4];
        const int4   dt = *(const int4*)(drow + (size_t)c * CHUNK_ELEMS);
        accum_elem(wh, qw, s.x, dt.x);
        accum_elem(wh, qw, s.y, dt.y);
        accum_elem(wh, qw, s.z, dt.z);
        accum_elem(wh, qw, s.w, dt.w);
    }

    __syncthreads();

    if (t < BINS) {
        unsigned tot = 0;
#pragma unroll
        for (int w = 0; w < NWAVE; ++w) tot += hcnt[w * 32 + t];
        out[(size_t)row * BINS + t] = __logf((float)tot + 1e-5f);
    }
}

extern "C" void kernel_launch(void* const* d_in, const int* in_sizes, int n_in,
                              void* d_out, int out_size, void* d_ws, size_t ws_size,
                              hipStream_t stream) {
    const float* simmat = (const float*)d_in[0];
    const int*   dtoks  = (const int*)d_in[1];
    const int*   qtoks  = (const int*)d_in[2];
    float*       out    = (float*)d_out;

    const int rows = in_sizes[0] / D_DIM;    // B * Q = 4096
    drmm_hist_kernel<<<rows, THREADS, 0, stream>>>(simmat, dtoks, qtoks, out);
}